// GinNet_90288802496750
// MI455X (gfx1250) — compile-verified
//
#include <hip/hip_runtime.h>
#include <hip/hip_bf16.h>

// ---------------------------------------------------------------------------
// GIN (3x GINConv + BN) on MI455X / gfx1250.
// Edge scatter-gather dominates (L2-atomic bound; feature maps fit in 192MB
// L2). GEMMs use FP32 WMMA (v_wmma_f32_16x16x4_f32), two fused per layer,
// with LDS weights pre-swizzled into exact B-fragment order so every
// fragment is a single ds_load_b64 feeding the WMMA directly.
// ---------------------------------------------------------------------------

#define N_NODES  200000
#define N_EDGES  3200000
#define N_GRAPHS 256
#define NH       64

typedef float v2f __attribute__((ext_vector_type(2)));
typedef float v8f __attribute__((ext_vector_type(8)));

static __device__ __forceinline__ v8f wmma_f32_k4(v2f a, v2f b, v8f c) {
    return __builtin_amdgcn_wmma_f32_16x16x4_f32(false, a, false, b, (short)0, c,
                                                 false, false);
}

// Map row-major weight element i = k*64+c to swizzled B-fragment slot:
//   frag = k/4, ntile = c/16, lane = ((k%4)/2)*16 + c%16, slot = k%2
//   index = (frag*4 + ntile)*64 + lane*2 + slot      (floats)
static __device__ __forceinline__ int swz_idx(int i) {
    int k = i >> 6, c = i & 63;
    int frag = k >> 2, rem = k & 3;
    int lane = ((rem >> 1) << 4) | (c & 15);
    int n = c >> 4;
    int slot = rem & 1;
    return (((frag << 2) | n) << 6) + (lane << 1) + slot;
}

// ---------------------------------------------------------------------------
// Edge aggregation, F=64: thread = (edge, 4-feature group).
// One global_load_b128 + 4 f32 atomic adds per thread.
// ---------------------------------------------------------------------------
__global__ __launch_bounds__(256) void k_agg64(const float* __restrict__ X,
                                               const int* __restrict__ src,
                                               const int* __restrict__ dst,
                                               float* __restrict__ AGG) {
    long long tid = (long long)blockIdx.x * blockDim.x + threadIdx.x;
    int e = (int)(tid >> 4);
    int q = (int)(tid & 15);          // 4-feature group
    int s = src[e];
    int d = dst[e];
    const float4 xv = *(const float4*)(X + (size_t)s * NH + q * 4);
    float* out = AGG + (size_t)d * NH + q * 4;
    atomicAdd(out + 0, xv.x);
    atomicAdd(out + 1, xv.y);
    atomicAdd(out + 2, xv.z);
    atomicAdd(out + 3, xv.w);
}

// Edge aggregation, F=2 (layer 1 input): thread = one edge.
__global__ __launch_bounds__(256) void k_agg2(const float* __restrict__ X,
                                              const int* __restrict__ src,
                                              const int* __restrict__ dst,
                                              float* __restrict__ AGG) {
    int e = blockIdx.x * blockDim.x + threadIdx.x;
    int s = src[e];
    int d = dst[e];
    const float2 xv = *(const float2*)(X + (size_t)s * 2);
    float* out = AGG + (size_t)d * 2;
    atomicAdd(out + 0, xv.x);
    atomicAdd(out + 1, xv.y);
}

// ---------------------------------------------------------------------------
// Layer-1 first linear (K=2, plain VALU): h1 = relu(((1+eps)x+agg) @ W + b)
// ---------------------------------------------------------------------------
__global__ __launch_bounds__(256) void k_lin1(const float* __restrict__ X,
                                              const float* __restrict__ AGG2,
                                              const float* __restrict__ epsP,
                                              const float* __restrict__ W,
                                              const float* __restrict__ b,
                                              float* __restrict__ OUT) {
    __shared__ float sW0[NH], sW1[NH], sb[NH];
    if (threadIdx.x < NH) {
        sW0[threadIdx.x] = W[threadIdx.x];
        sW1[threadIdx.x] = W[NH + threadIdx.x];
        sb[threadIdx.x]  = b[threadIdx.x];
    }
    __syncthreads();
    int tid  = blockIdx.x * blockDim.x + threadIdx.x;   // node*64 + j
    int node = tid >> 6;
    int j    = tid & 63;
    float epv = 1.0f + epsP[0];
    const float2 xv = *(const float2*)(X + (size_t)node * 2);
    const float2 av = *(const float2*)(AGG2 + (size_t)node * 2);
    float z0 = fmaf(epv, xv.x, av.x);
    float z1 = fmaf(epv, xv.y, av.y);
    float v  = fmaf(z0, sW0[j], fmaf(z1, sW1[j], sb[j]));
    OUT[tid] = v > 0.0f ? v : 0.0f;
}

// ---------------------------------------------------------------------------
// Single 64x64 GEMM via FP32 WMMA:  OUT = relu(IN @ W + b), + BN stats.
// 128 threads = 4 waves; wave handles 16 rows x 64 cols; grid = 3125.
// ---------------------------------------------------------------------------
__global__ __launch_bounds__(128) void k_gemm_relu_stats(
    const float* __restrict__ IN, const float* __restrict__ W,
    const float* __restrict__ b, float* __restrict__ OUT,
    float* __restrict__ colsum, float* __restrict__ colsq) {
    __shared__ float sW[NH * NH];     // swizzled B-fragment layout
    __shared__ float sb[NH];

    int t = threadIdx.x;
    for (int i = t; i < NH * NH; i += 128) sW[swz_idx(i)] = W[i];
    if (t < NH) sb[t] = b[t];
    __syncthreads();

    int wave  = t >> 5;
    int lane  = t & 31;
    int mrow  = lane & 15;
    int khalf = (lane >> 4) * 2;          // 0 or 2
    int row0  = blockIdx.x * 64 + wave * 16;

    const float* Irow = IN + (size_t)(row0 + mrow) * NH;
    const float2* Wfrag = (const float2*)sW;

    v8f acc[4];
#pragma unroll
    for (int n = 0; n < 4; ++n)
#pragma unroll
        for (int r = 0; r < 8; ++r) acc[n][r] = 0.0f;

    for (int f = 0; f < 16; ++f) {
        int kk = f * 4 + khalf;
        float2 iv = *(const float2*)(Irow + kk);
        v2f a; a.x = iv.x; a.y = iv.y;
#pragma unroll
        for (int n = 0; n < 4; ++n) {
            float2 bp = Wfrag[(f * 4 + n) * 32 + lane];
            v2f bb; bb.x = bp.x; bb.y = bp.y;
            acc[n] = wmma_f32_k4(a, bb, acc[n]);
        }
    }

#pragma unroll
    for (int n = 0; n < 4; ++n) {
        int col = n * 16 + mrow;
        float bv = sb[col];
        float ps = 0.0f, pq = 0.0f;
#pragma unroll
        for (int r = 0; r < 8; ++r) {
            int m = r + 8 * (lane >> 4);
            float v = acc[n][r] + bv;
            v = v > 0.0f ? v : 0.0f;
            OUT[(size_t)(row0 + m) * NH + col] = v;
            ps += v;
            pq += v * v;
        }
        atomicAdd(&colsum[col], ps);
        atomicAdd(&colsq[col], pq);
    }
}

// ---------------------------------------------------------------------------
// Fused GIN MLP for layers 2/3:
//   Z = (1+eps)*X + AGG;  H1 = relu(Z@Wa + ba);  OUT = relu(H1@Wb + bb)
// H1 staged per-wave in LDS to re-layout C/D fragments -> A fragments.
// Also accumulates BN column sums / sums-of-squares.
// ---------------------------------------------------------------------------
__global__ __launch_bounds__(128) void k_gin_fused(
    const float* __restrict__ X, const float* __restrict__ AGG,
    const float* __restrict__ epsP, const float* __restrict__ Wa,
    const float* __restrict__ ba, const float* __restrict__ Wb,
    const float* __restrict__ bb, float* __restrict__ OUT,
    float* __restrict__ colsum, float* __restrict__ colsq) {
    __shared__ float sWa[NH * NH];    // swizzled B-fragment layout
    __shared__ float sWb[NH * NH];    // swizzled B-fragment layout
    __shared__ float sba[NH];
    __shared__ float sbb[NH];
    __shared__ float sH1[4][16 * 68]; // per-wave staging, padded stride 68

    int t = threadIdx.x;
    for (int i = t; i < NH * NH; i += 128) {
        int si = swz_idx(i);
        sWa[si] = Wa[i];
        sWb[si] = Wb[i];
    }
    if (t < NH) {
        sba[t] = ba[t];
        sbb[t] = bb[t];
    }
    __syncthreads();

    float epv = 1.0f + epsP[0];
    int wave  = t >> 5;
    int lane  = t & 31;
    int mrow  = lane & 15;
    int khalf = (lane >> 4) * 2;
    int row0  = blockIdx.x * 64 + wave * 16;

    const float* Xrow = X + (size_t)(row0 + mrow) * NH;
    const float* Arow = AGG + (size_t)(row0 + mrow) * NH;
    const float2* WaFrag = (const float2*)sWa;
    const float2* WbFrag = (const float2*)sWb;

    // ---- GEMM1: Z @ Wa ----
    v8f acc[4];
#pragma unroll
    for (int n = 0; n < 4; ++n)
#pragma unroll
        for (int r = 0; r < 8; ++r) acc[n][r] = 0.0f;

    for (int f = 0; f < 16; ++f) {
        int kk = f * 4 + khalf;
        float2 xv = *(const float2*)(Xrow + kk);
        float2 av = *(const float2*)(Arow + kk);
        v2f a;
        a.x = fmaf(epv, xv.x, av.x);
        a.y = fmaf(epv, xv.y, av.y);
#pragma unroll
        for (int n = 0; n < 4; ++n) {
            float2 bp = WaFrag[(f * 4 + n) * 32 + lane];
            v2f b; b.x = bp.x; b.y = bp.y;
            acc[n] = wmma_f32_k4(a, b, acc[n]);
        }
    }

    // H1 = relu(acc + ba): C/D layout -> LDS (row-major per wave)
    float* H = &sH1[wave][0];
#pragma unroll
    for (int n = 0; n < 4; ++n) {
        int col = n * 16 + mrow;
        float bv = sba[col];
#pragma unroll
        for (int r = 0; r < 8; ++r) {
            int m = r + 8 * (lane >> 4);
            float v = acc[n][r] + bv;
            H[m * 68 + col] = v > 0.0f ? v : 0.0f;
        }
    }
    // Same-wave LDS RAW: DS ops are in-order; compiler inserts s_wait_dscnt.

    // ---- GEMM2: H1 @ Wb ----
    v8f d[4];
#pragma unroll
    for (int n = 0; n < 4; ++n)
#pragma unroll
        for (int r = 0; r < 8; ++r) d[n][r] = 0.0f;

    for (int f = 0; f < 16; ++f) {
        int kk = f * 4 + khalf;
        // 272-byte row stride keeps this 8B aligned -> single ds_load_b64
        float2 hv = *(const float2*)(H + mrow * 68 + kk);
        v2f a; a.x = hv.x; a.y = hv.y;
#pragma unroll
        for (int n = 0; n < 4; ++n) {
            float2 bp = WbFrag[(f * 4 + n) * 32 + lane];
            v2f b; b.x = bp.x; b.y = bp.y;
            d[n] = wmma_f32_k4(a, b, d[n]);
        }
    }

    // OUT = relu(d + bb) + BN stats
#pragma unroll
    for (int n = 0; n < 4; ++n) {
        int col = n * 16 + mrow;
        float bv = sbb[col];
        float ps = 0.0f, pq = 0.0f;
#pragma unroll
        for (int r = 0; r < 8; ++r) {
            int m = r + 8 * (lane >> 4);
            float v = d[n][r] + bv;
            v = v > 0.0f ? v : 0.0f;
            OUT[(size_t)(row0 + m) * NH + col] = v;
            ps += v;
            pq += v * v;
        }
        atomicAdd(&colsum[col], ps);
        atomicAdd(&colsq[col], pq);
    }
}

// ---------------------------------------------------------------------------
// BatchNorm apply (in place): y = g*(x-m)*rsqrt(v+eps)+be, stats from sums.
// ---------------------------------------------------------------------------
__global__ __launch_bounds__(256) void k_bn(float* __restrict__ H,
                                            const float* __restrict__ colsum,
                                            const float* __restrict__ colsq,
                                            const float* __restrict__ g,
                                            const float* __restrict__ be) {
    __shared__ float sc[NH], sh[NH];
    if (threadIdx.x < NH) {
        float invN = 1.0f / (float)N_NODES;
        float m = colsum[threadIdx.x] * invN;
        float v = colsq[threadIdx.x] * invN - m * m;
        float s = g[threadIdx.x] * rsqrtf(v + 1e-5f);
        sc[threadIdx.x] = s;
        sh[threadIdx.x] = be[threadIdx.x] - m * s;
    }
    __syncthreads();
    size_t nvec   = (size_t)N_NODES * NH / 4;   // float4 count
    size_t stride = (size_t)gridDim.x * blockDim.x;
    for (size_t i = (size_t)blockIdx.x * blockDim.x + threadIdx.x; i < nvec;
         i += stride) {
        float4 h = ((float4*)H)[i];
        int c = (int)((i * 4) & 63);
        h.x = fmaf(h.x, sc[c + 0], sh[c + 0]);
        h.y = fmaf(h.y, sc[c + 1], sh[c + 1]);
        h.z = fmaf(h.z, sc[c + 2], sh[c + 2]);
        h.w = fmaf(h.w, sc[c + 3], sh[c + 3]);
        ((float4*)H)[i] = h;
    }
}

// ---------------------------------------------------------------------------
// Global mean pool (accumulate): pooled[batch[n]] += h[n]; cnt[batch[n]] += 1
// ---------------------------------------------------------------------------
__global__ __launch_bounds__(256) void k_pool(const float* __restrict__ H,
                                              const int* __restrict__ batch,
                                              float* __restrict__ pooled,
                                              float* __restrict__ cnt) {
    int tid  = blockIdx.x * blockDim.x + threadIdx.x;   // node*64 + c
    int node = tid >> 6;
    int c    = tid & 63;
    int gph  = batch[node];
    atomicAdd(&pooled[(size_t)gph * NH + c], H[tid]);
    if (c == 0) atomicAdd(&cnt[gph], 1.0f);
}

// ---------------------------------------------------------------------------
// Head: mean, ELU(pooled@Wf1+bf1), log_softmax((..)@Wf2+bf2). 1 block/graph.
// ---------------------------------------------------------------------------
__global__ __launch_bounds__(64) void k_head(const float* __restrict__ pooled,
                                             const float* __restrict__ cnt,
                                             const float* __restrict__ Wf1,
                                             const float* __restrict__ bf1,
                                             const float* __restrict__ Wf2,
                                             const float* __restrict__ bf2,
                                             float* __restrict__ out) {
    __shared__ float p[NH], z[32], o[10], lse[1];
    int g = blockIdx.x, t = threadIdx.x;
    float c = cnt[g];
    c = c > 1.0f ? c : 1.0f;
    p[t] = pooled[(size_t)g * NH + t] / c;
    __syncthreads();
    if (t < 32) {
        float a = bf1[t];
        for (int k = 0; k < NH; ++k) a = fmaf(p[k], Wf1[k * 32 + t], a);
        z[t] = a > 0.0f ? a : (expf(a) - 1.0f);   // ELU(alpha=1)
    }
    __syncthreads();
    if (t < 10) {
        float a = bf2[t];
        for (int k = 0; k < 32; ++k) a = fmaf(z[k], Wf2[k * 10 + t], a);
        o[t] = a;
    }
    __syncthreads();
    if (t == 0) {
        float mx = o[0];
        for (int j = 1; j < 10; ++j) mx = fmaxf(mx, o[j]);
        float se = 0.0f;
        for (int j = 0; j < 10; ++j) se += expf(o[j] - mx);
        lse[0] = mx + logf(se);
    }
    __syncthreads();
    if (t < 10) out[(size_t)g * 10 + t] = o[t] - lse[0];
}

// ---------------------------------------------------------------------------
// Host launcher
// ---------------------------------------------------------------------------
extern "C" void kernel_launch(void* const* d_in, const int* in_sizes, int n_in,
                              void* d_out, int out_size, void* d_ws,
                              size_t ws_size, hipStream_t stream) {
    const float* x     = (const float*)d_in[0];
    const int*   ei    = (const int*)d_in[1];
    const int*   batch = (const int*)d_in[2];
    const float* eps1  = (const float*)d_in[3];
    const float* W1a   = (const float*)d_in[4];
    const float* b1a   = (const float*)d_in[5];
    const float* W1b   = (const float*)d_in[6];
    const float* b1b   = (const float*)d_in[7];
    const float* g1    = (const float*)d_in[8];
    const float* be1   = (const float*)d_in[9];
    const float* eps2  = (const float*)d_in[10];
    const float* W2a   = (const float*)d_in[11];
    const float* b2a   = (const float*)d_in[12];
    const float* W2b   = (const float*)d_in[13];
    const float* b2b   = (const float*)d_in[14];
    const float* g2    = (const float*)d_in[15];
    const float* be2   = (const float*)d_in[16];
    const float* eps3  = (const float*)d_in[17];
    const float* W3a   = (const float*)d_in[18];
    const float* b3a   = (const float*)d_in[19];
    const float* W3b   = (const float*)d_in[20];
    const float* b3b   = (const float*)d_in[21];
    const float* g3    = (const float*)d_in[22];
    const float* be3   = (const float*)d_in[23];
    const float* Wf1   = (const float*)d_in[24];
    const float* bf1   = (const float*)d_in[25];
    const float* Wf2   = (const float*)d_in[26];
    const float* bf2   = (const float*)d_in[27];

    const int* src = ei;
    const int* dst = ei + N_EDGES;

    const size_t FEAT_BYTES = (size_t)N_NODES * NH * sizeof(float);  // 51.2 MB
    char* ws = (char*)d_ws;
    float* bufA   = (float*)(ws);
    float* bufB   = (float*)(ws + FEAT_BYTES);
    float* agg    = (float*)(ws + 2 * FEAT_BYTES);
    float* colsum = (float*)(ws + 3 * FEAT_BYTES);
    float* colsq  = colsum + NH;
    float* pooled = colsum + 2 * NH;
    float* cnt    = pooled + N_GRAPHS * NH;

    const int GEMM_GRID  = N_NODES / 64;                        // 3125 (exact)
    const int ELEM_GRID  = N_NODES * NH / 256;                  // 50000 (exact)
    const int AGG64_GRID = (int)((size_t)N_EDGES * 16 / 256);   // 200000
    const int AGG2_GRID  = N_EDGES / 256;                       // 12500

    // ---------------- Layer 1 ----------------
    hipMemsetAsync(agg, 0, (size_t)N_NODES * 2 * sizeof(float), stream);
    k_agg2<<<AGG2_GRID, 256, 0, stream>>>(x, src, dst, agg);
    k_lin1<<<ELEM_GRID, 256, 0, stream>>>(x, agg, eps1, W1a, b1a, bufB);
    hipMemsetAsync(colsum, 0, 2 * NH * sizeof(float), stream);
    k_gemm_relu_stats<<<GEMM_GRID, 128, 0, stream>>>(bufB, W1b, b1b, bufA,
                                                     colsum, colsq);
    k_bn<<<3200, 256, 0, stream>>>(bufA, colsum, colsq, g1, be1);

    // ---------------- Layer 2 ----------------
    hipMemsetAsync(agg, 0, FEAT_BYTES, stream);
    k_agg64<<<AGG64_GRID, 256, 0, stream>>>(bufA, src, dst, agg);
    hipMemsetAsync(colsum, 0, 2 * NH * sizeof(float), stream);
    k_gin_fused<<<GEMM_GRID, 128, 0, stream>>>(bufA, agg, eps2, W2a, b2a, W2b,
                                               b2b, bufB, colsum, colsq);
    k_bn<<<3200, 256, 0, stream>>>(bufB, colsum, colsq, g2, be2);

    // ---------------- Layer 3 ----------------
    hipMemsetAsync(agg, 0, FEAT_BYTES, stream);
    k_agg64<<<AGG64_GRID, 256, 0, stream>>>(bufB, src, dst, agg);
    hipMemsetAsync(colsum, 0, 2 * NH * sizeof(float), stream);
    k_gin_fused<<<GEMM_GRID, 128, 0, stream>>>(bufB, agg, eps3, W3a, b3a, W3b,
                                               b3b, bufA, colsum, colsq);
    k_bn<<<3200, 256, 0, stream>>>(bufA, colsum, colsq, g3, be3);

    // ---------------- Pool + head ----------------
    hipMemsetAsync(pooled, 0, (size_t)(N_GRAPHS * NH + N_GRAPHS) * sizeof(float),
                   stream);
    k_pool<<<ELEM_GRID, 256, 0, stream>>>(bufA, batch, pooled, cnt);
    k_head<<<N_GRAPHS, 64, 0, stream>>>(pooled, cnt, Wf1, bf1, Wf2, bf2,
                                        (float*)d_out);
}